// Subtraction_14061722927493
// MI455X (gfx1250) — compile-verified
//
#include <hip/hip_runtime.h>

// ---------------------------------------------------------------------------
// subtraction op: out[n,c,kk,oh*96+ow] = x[n,c,oh,ow] - xpad[n,c,oh+i,ow+j]
// x: (8,32,96,96) f32, K=7, S=1, P=3, D=1, reflect padding.
// Store-bandwidth bound: 462 MB out vs 9.4 MB in  ->  ~20 us floor @ 23.3 TB/s.
// CDNA5 path: per-plane reflect-padded tile staged to LDS with
// GLOBAL_LOAD_ASYNC_TO_LDS_B32 (per-lane global->LDS gather), synchronized
// with s_wait_asynccnt; output written with non-temporal 128-bit stores.
// ---------------------------------------------------------------------------

typedef float f32x4 __attribute__((ext_vector_type(4)));

// Builtin parameter types (from hipcc diagnostic): AS(1) int* and AS(3) int*
typedef __attribute__((address_space(1))) int g_int;
typedef __attribute__((address_space(3))) int l_int;

#define N_PLANES 256   // 8 * 32
#define H        96
#define W        96
#define PIX      (H * W)      // 9216
#define KDIM     7
#define KK       49
#define PADW     102          // 96 + 2*3
#define PITCH    104          // padded row pitch (floats) to stagger LDS banks
#define THREADS  256          // 8 wave32 waves

__device__ __forceinline__ int reflect96(int t) {
    // numpy 'reflect' (no edge repeat), valid for t in [-3, 98]
    if (t < 0)  t = -t;
    if (t > 95) t = 190 - t;
    return t;
}

__global__ __launch_bounds__(THREADS)
void Subtraction_14061722927493_kernel(const float* __restrict__ x,
                                       float* __restrict__ out) {
    __shared__ float sp[PADW * PITCH];   // 102 * 104 * 4 B = 42.4 KB of 320 KB

    const int plane = blockIdx.x;        // 0..255 -> (n,c)
    const int part  = blockIdx.y;        // 0..1   -> tap split
    const int tid   = threadIdx.x;

    const float* __restrict__ xp = x + (size_t)plane * PIX;

    // ---- stage reflect-padded plane into LDS (async global->LDS gather) ----
    for (int idx = tid; idx < PADW * PADW; idx += THREADS) {
        const int pr = idx / PADW;
        const int pc = idx - pr * PADW;
        const int sr = reflect96(pr - 3);
        const int sc = reflect96(pc - 3);
        const float* gp = xp + sr * W + sc;
        float*       lp = &sp[pr * PITCH + pc];
#if __has_builtin(__builtin_amdgcn_global_load_async_to_lds_b32)
        __builtin_amdgcn_global_load_async_to_lds_b32(
            (g_int*)const_cast<float*>(gp),
            (l_int*)lp,
            /*offset=*/0, /*cpol=*/0);
#else
        *lp = *gp;   // fallback: plain load/store through VGPRs
#endif
    }
#if __has_builtin(__builtin_amdgcn_global_load_async_to_lds_b32)
#if __has_builtin(__builtin_amdgcn_s_wait_asynccnt)
    __builtin_amdgcn_s_wait_asynccnt(0);
#else
    asm volatile("s_wait_asynccnt 0" ::: "memory");
#endif
#endif
    __syncthreads();

    // ---- compute: pixels outer (center in regs), taps inner, NT stores ----
    float* __restrict__ oplane = out + (size_t)plane * (size_t)KK * PIX;

    // 9216 pixels / 4 per chunk = 2304 chunks; 2304 / 256 threads = 9 iters
    for (int p4 = tid; p4 < PIX / 4; p4 += THREADS) {
        const int oh = p4 / (W / 4);
        const int ow = (p4 - oh * (W / 4)) * 4;

        const float* c = &sp[(oh + 3) * PITCH + (ow + 3)];
        const float c0 = c[0], c1 = c[1], c2 = c[2], c3 = c[3];

        for (int kk = part; kk < KK; kk += 2) {
            const int i = kk / KDIM;
            const int j = kk - i * KDIM;
            const float* nb = &sp[(oh + i) * PITCH + (ow + j)];
            f32x4 v;
            v.x = c0 - nb[0];
            v.y = c1 - nb[1];
            v.z = c2 - nb[2];
            v.w = c3 - nb[3];
            float* dst = oplane + (size_t)kk * PIX + (size_t)p4 * 4;
            __builtin_nontemporal_store(v, (f32x4*)dst);
        }
    }
}

extern "C" void kernel_launch(void* const* d_in, const int* in_sizes, int n_in,
                              void* d_out, int out_size, void* d_ws, size_t ws_size,
                              hipStream_t stream) {
    (void)in_sizes; (void)n_in; (void)out_size; (void)d_ws; (void)ws_size;
    const float* x = (const float*)d_in[0];
    float* out     = (float*)d_out;

    dim3 grid(N_PLANES, 2);   // 256 (n,c) planes x 2 tap-halves = 512 blocks
    dim3 block(THREADS);
    Subtraction_14061722927493_kernel<<<grid, block, 0, stream>>>(x, out);
}